// gradmap_55259049230648
// MI455X (gfx1250) — compile-verified
//
#include <hip/hip_runtime.h>
#include <stdint.h>

// ---------------------------------------------------------------------------
// Problem geometry
// ---------------------------------------------------------------------------
#define IMG_H 1024
#define IMG_W 1024
#define NIMG  12
#define SENT  (IMG_H * IMG_W)          // sentinel label (h*w)
#define CSTRIDE (IMG_H * IMG_W + 16)   // per-image counts stride (>= h*w+1)
static constexpr size_t IMG = (size_t)IMG_H * (size_t)IMG_W;

static constexpr float T_HALF = (float)(0.5 / 1023.0);
static constexpr float T_FOUR = (float)(4.0 / 1023.0);
static constexpr float T_ONE  = (float)(1.0 / 1023.0);
static constexpr float SCALE  = (float)(1023.0 / 4.0);

// ---------------------------------------------------------------------------
// CDNA5 async global->LDS copies (gfx1250 async data path, ASYNCcnt-tracked).
// VDST = LDS byte address (low 32 bits of the generic LDS pointer == AS(3)
// offset per the CDNA5 aperture rules), VADDR = 64-bit global address.
// b128 form moves 16 B/lane -> full-cacheline requests, 4x fewer instructions.
// ---------------------------------------------------------------------------
__device__ __forceinline__ void async_ld_b128(const void* lds_dst, const void* gsrc) {
  uint32_t loff = (uint32_t)(uintptr_t)lds_dst;
  asm volatile("global_load_async_to_lds_b128 %0, %1, off"
               :: "v"(loff), "v"(gsrc) : "memory");
}
__device__ __forceinline__ void async_wait0() {
#if __has_builtin(__builtin_amdgcn_s_wait_asynccnt)
  __builtin_amdgcn_s_wait_asynccnt(0);
#else
  asm volatile("s_wait_asynccnt 0" ::: "memory");
#endif
}

__device__ __forceinline__ int refl(int i) {       // jnp.pad mode='reflect', pad=1
  return i < 0 ? -i : (i > IMG_H - 1 ? 2 * (IMG_H - 1) - i : i);
}

// ---------------------------------------------------------------------------
// Stage 1: gradmask + label init (fused).  gm = max(gx,gy)*1023/4,
// lab0 = fg ? r*W+c : SENT  with fg = gm > 1.
// Interior blocks stage a 22x24 f32 tile (radius-3 halo, 16B-aligned columns
// c0-4 .. c0+19) with one round of async b128 copies.
// ---------------------------------------------------------------------------
__global__ void __launch_bounds__(256)
k_grad(const float* __restrict__ img, float* __restrict__ gm, int* __restrict__ lab) {
  __shared__ alignas(16) float S[22][24];
  const int tx = threadIdx.x, ty = threadIdx.y;
  const int bx = blockIdx.x,  by = blockIdx.y, z = blockIdx.z;
  const int c0 = bx * 16, r0 = by * 16;
  const int t  = ty * 16 + tx;
  const float* im = img + (size_t)z * IMG;

  const bool interior = (bx >= 1) && (bx <= 62) && (by >= 1) && (by <= 62);
  if (interior) {
    int e = t; if (e > 131) e = 131;           // 22 rows x 6 b128-quads = 132
    int er = e / 6, eq = e % 6;
    async_ld_b128(&S[er][eq * 4], im + (r0 - 3 + er) * IMG_W + (c0 - 4 + eq * 4));
    async_wait0();
  } else {
    for (int e = t; e < 22 * 24; e += 256) {
      int er = e / 24, ec = e % 24;
      int gr = r0 - 3 + er, gc = c0 - 4 + ec;
      S[er][ec] = (gr >= 0 && gr < IMG_H && gc >= 0 && gc < IMG_W)
                      ? im[gr * IMG_W + gc] : 0.0f;   // zero-fill == _shift pad
    }
  }
  __syncthreads();

  auto ld = [&](int dr, int dc) -> float { return S[ty + 3 + dr][tx + 4 + dc]; };

  float gx, gy;
  {
    float g1 = ld(0, -1) - ld(0, 1);
    float g2 = ld(0, -2) - ld(0, 2);
    float g3 = ld(0, -3) - ld(0, 3);
    float gmrg = (g1 + g2 + g3) / 3.0f;
    float a = fabsf(g1);
    bool keep = (fabsf(gmrg - g1) <= T_HALF) && (a <= T_FOUR) && (a > T_ONE);
    gx = keep ? a : 0.0f;
  }
  {
    float g1 = ld(-1, 0) - ld(1, 0);
    float g2 = ld(-2, 0) - ld(2, 0);
    float g3 = ld(-3, 0) - ld(3, 0);
    float gmrg = (g1 + g2 + g3) / 3.0f;
    float a = fabsf(g1);
    bool keep = (fabsf(gmrg - g1) <= T_HALF) && (a <= T_FOUR) && (a > T_ONE);
    gy = keep ? a : 0.0f;
  }
  const int r = r0 + ty, c = c0 + tx;
  float g = fmaxf(gx, gy) * SCALE;
  size_t p = (size_t)z * IMG + (size_t)r * IMG_W + (size_t)c;
  gm[p]  = g;
  lab[p] = (g > 1.0f) ? (r * IMG_W + c) : SENT;
}

// ---------------------------------------------------------------------------
// Zero the counts region (workspace is poisoned / stale between calls).
// ---------------------------------------------------------------------------
__global__ void __launch_bounds__(256)
k_zero(float* __restrict__ counts, int n) {
  int i = blockIdx.x * 256 + threadIdx.x;
  if (i < n) counts[i] = 0.0f;
}

// ---------------------------------------------------------------------------
// Stage 2: one ping-pong pass of 8-connected min-label propagation.
// 18x24 label tile+halo (cols c0-4 .. c0+19, 16B-aligned) staged with one
// round of async b128 copies for interior blocks; Jacobi-iterate to a local
// fixpoint.  Reads labIn, writes own 16x16 interior to labOut -> deterministic.
// ---------------------------------------------------------------------------
__global__ void __launch_bounds__(256)
k_prop(const int* __restrict__ labIn, int* __restrict__ labOut) {
  __shared__ alignas(16) int T[18][24];
  const int tx = threadIdx.x, ty = threadIdx.y;
  const int bx = blockIdx.x,  by = blockIdx.y, z = blockIdx.z;
  const int c0 = bx * 16, r0 = by * 16;
  const int t = ty * 16 + tx;
  const int* Lin  = labIn  + (size_t)z * IMG;
  int*       Lout = labOut + (size_t)z * IMG;

  const bool interior = (bx >= 1) && (bx <= 62) && (by >= 1) && (by <= 62);
  if (interior) {
    int e = t; if (e > 107) e = 107;           // 18 rows x 6 b128-quads = 108
    int er = e / 6, eq = e % 6;
    async_ld_b128(&T[er][eq * 4], Lin + (r0 - 1 + er) * IMG_W + (c0 - 4 + eq * 4));
    async_wait0();
  } else {
    for (int e = t; e < 18 * 24; e += 256) {
      int er = e / 24, ec = e % 24;
      int gr = r0 - 1 + er, gc = c0 - 4 + ec;
      T[er][ec] = (gr >= 0 && gr < IMG_H && gc >= 0 && gc < IMG_W)
                      ? Lin[gr * IMG_W + gc] : SENT;
    }
  }
  __syncthreads();

  // local coords: row = ty+1 (halo 1), col = tx+4 (halo 4 due to alignment)
  int my = T[ty + 1][tx + 4];
  for (int it = 0; it < 24; ++it) {
    int nv = my;
    if (my != SENT) {
      nv = min(nv, T[ty    ][tx + 3]);
      nv = min(nv, T[ty    ][tx + 4]);
      nv = min(nv, T[ty    ][tx + 5]);
      nv = min(nv, T[ty + 1][tx + 3]);
      nv = min(nv, T[ty + 1][tx + 5]);
      nv = min(nv, T[ty + 2][tx + 3]);
      nv = min(nv, T[ty + 2][tx + 4]);
      nv = min(nv, T[ty + 2][tx + 5]);
    }
    __syncthreads();                   // all reads done before writes (Jacobi)
    int ch = (nv < my) ? 1 : 0;
    if (ch) { T[ty + 1][tx + 4] = nv; my = nv; }
    if (__syncthreads_count(ch) == 0) break;
  }
  Lout[(r0 + ty) * IMG_W + (c0 + tx)] = my;
}

// ---------------------------------------------------------------------------
// Stage 3: per-component boundary-pixel counts (exact: sums of 1.0f, so the
// float atomics are order-independent -> deterministic).
// ---------------------------------------------------------------------------
__global__ void __launch_bounds__(256)
k_count(const float* __restrict__ gm, const int* __restrict__ lab,
        float* __restrict__ counts) {
  const int c = blockIdx.x * 16 + threadIdx.x;
  const int r = blockIdx.y * 16 + threadIdx.y;
  const int z = blockIdx.z;
  const float* g = gm + (size_t)z * IMG;
  float gv = g[r * IMG_W + c];
  if (gv > 1.0f) {
    bool in4 = (r > 0         && g[(r - 1) * IMG_W + c] > 1.0f) &&
               (r < IMG_H - 1 && g[(r + 1) * IMG_W + c] > 1.0f) &&
               (c > 0         && g[r * IMG_W + c - 1]   > 1.0f) &&
               (c < IMG_W - 1 && g[r * IMG_W + c + 1]   > 1.0f);
    if (!in4) {
      int lv = lab[(size_t)z * IMG + (size_t)r * IMG_W + c];
      atomicAdd(&counts[(size_t)z * CSTRIDE + lv], 1.0f);
    }
  }
}

// ---------------------------------------------------------------------------
// Stage 4: directmask from gm (LDS-tiled, reflect-padded 3x3 blur of the
// structure tensor), lengthmask lookup, final clip.
// 20x24 gm tile (radius-2 halo, aligned cols) via one async b128 round.
// ---------------------------------------------------------------------------
__global__ void __launch_bounds__(256)
k_final(const float* __restrict__ gm, const int* __restrict__ lab,
        const float* __restrict__ counts, float* __restrict__ out) {
  __shared__ alignas(16) float G[20][24];   // gm tile, zero-filled OOB
  __shared__ float GX[18][19];              // gx tile, halo 1 (padded stride)
  __shared__ float GY[18][19];              // gy tile, halo 1
  const int tx = threadIdx.x, ty = threadIdx.y;
  const int bx = blockIdx.x,  by = blockIdx.y, z = blockIdx.z;
  const int c0 = bx * 16, r0 = by * 16;
  const int t = ty * 16 + tx;
  const float* gmi = gm + (size_t)z * IMG;

  const bool interior = (bx >= 1) && (bx <= 62) && (by >= 1) && (by <= 62);
  if (interior) {
    int e = t; if (e > 119) e = 119;         // 20 rows x 6 b128-quads = 120
    int er = e / 6, eq = e % 6;
    async_ld_b128(&G[er][eq * 4], gmi + (r0 - 2 + er) * IMG_W + (c0 - 4 + eq * 4));
    async_wait0();
  } else {
    for (int e = t; e < 20 * 24; e += 256) {
      int er = e / 24, ec = e % 24;
      int gr = r0 - 2 + er, gc = c0 - 4 + ec;
      G[er][ec] = (gr >= 0 && gr < IMG_H && gc >= 0 && gc < IMG_W)
                      ? gmi[gr * IMG_W + gc] : 0.0f;
    }
  }
  __syncthreads();

  // gx(R,C) = gmZ(R,C-1) - gmZ(R,C+1); gy(R,C) = gmZ(R-1,C) - gmZ(R+1,C)
  // GX/GY local (er,ec) <-> global (r0-1+er, c0-1+ec);  G[a][b] = gm(r0-2+a, c0-4+b)
  for (int e = t; e < 18 * 18; e += 256) {
    int er = e / 18, ec = e % 18;
    GX[er][ec] = G[er + 1][ec + 2] - G[er + 1][ec + 4];
    GY[er][ec] = G[er][ec + 3]     - G[er + 2][ec + 3];
  }
  __syncthreads();

  const int r = r0 + ty, c = c0 + tx;
  float sxx = 0.0f, syy = 0.0f, sxy = 0.0f;
#pragma unroll
  for (int dy = -1; dy <= 1; ++dy) {
    int lr = refl(r + dy) - (r0 - 1);     // stays in [0,17] incl. reflected cases
#pragma unroll
    for (int dx = -1; dx <= 1; ++dx) {
      int lc = refl(c + dx) - (c0 - 1);
      float vx = GX[lr][lc];
      float vy = GY[lr][lc];
      sxx += vx * vx;
      syy += vy * vy;
      sxy += vx * vy;
    }
  }
  sxx /= 9.0f; syy /= 9.0f; sxy /= 9.0f;
  float d   = sxx - syy;
  float tmp = sqrtf(d * d + 4.0f * sxy * sxy);
  float dm  = tmp / (sxx + syy + 1e-16f);

  float g  = G[ty + 2][tx + 4];
  bool fg  = g > 1.0f;
  bool in4 = (G[ty + 1][tx + 4] > 1.0f) && (G[ty + 3][tx + 4] > 1.0f) &&
             (G[ty + 2][tx + 3] > 1.0f) && (G[ty + 2][tx + 5] > 1.0f);
  bool bnd = fg && !in4;

  size_t p = (size_t)z * IMG + (size_t)r * IMG_W + (size_t)c;
  float lm = 0.0f;
  if (bnd) {
    int lv = lab[p];
    float cv = counts[(size_t)z * CSTRIDE + lv];
    if (cv > 21.0f) lm = cv;
  }
  out[p] = fminf(fmaxf(g * dm * lm, 0.0f), 1.0f);
}

// ---------------------------------------------------------------------------
// Launch: grad+init -> zero counts -> 24 ping-pong propagation passes
// (even count -> result in labA) -> boundary counts -> final.
// ---------------------------------------------------------------------------
extern "C" void kernel_launch(void* const* d_in, const int* in_sizes, int n_in,
                              void* d_out, int out_size, void* d_ws, size_t ws_size,
                              hipStream_t stream) {
  (void)in_sizes; (void)n_in; (void)out_size; (void)ws_size;
  const float* base = (const float*)d_in[0];
  float* out = (float*)d_out;

  char* ws = (char*)d_ws;
  float* gmBuf   = (float*)(ws);                       // 12M f32  (48 MB)
  int*   labA    = (int*)  (ws + 12 * IMG * 4);        // 12M i32  (48 MB)
  int*   labB    = (int*)  (ws + 24 * IMG * 4);        // 12M i32  (48 MB)
  float* counts  = (float*)(ws + 36 * IMG * 4);        // 12*CSTRIDE f32 (~50 MB)

  dim3 blk(16, 16, 1);
  dim3 grd(IMG_W / 16, IMG_H / 16, NIMG);

  k_grad<<<grd, blk, 0, stream>>>(base, gmBuf, labA);

  int ncnt = NIMG * CSTRIDE;
  k_zero<<<(ncnt + 255) / 256, 256, 0, stream>>>(counts, ncnt);

  const int PASSES = 24;                               // even -> result in labA
  for (int i = 0; i < PASSES; ++i) {
    int* lin  = (i & 1) ? labB : labA;
    int* lout = (i & 1) ? labA : labB;
    k_prop<<<grd, blk, 0, stream>>>(lin, lout);
  }

  k_count<<<grd, blk, 0, stream>>>(gmBuf, labA, counts);
  k_final<<<grd, blk, 0, stream>>>(gmBuf, labA, counts, out);
}